// Mnist_graph_pred_GNN_56667798504114
// MI455X (gfx1250) — compile-verified
//
#include <hip/hip_runtime.h>
#include <hip/hip_bf16.h>
#include <math.h>

typedef __attribute__((ext_vector_type(2))) float v2f;
typedef __attribute__((ext_vector_type(8))) float v8f;

// ---------------------------------------------------------------------------
// 1) deg[i] = 1 (self loop)
__global__ void k_init_deg(unsigned* __restrict__ deg, int n) {
    int i = blockIdx.x * blockDim.x + threadIdx.x;
    if (i < n) deg[i] = 1u;
}

// 2) integer degree count over dst (exact + deterministic)
__global__ void k_count_deg(const int* __restrict__ dst, unsigned* __restrict__ deg, int e) {
    int i = blockIdx.x * blockDim.x + threadIdx.x;
    if (i < e) atomicAdd(&deg[dst[i]], 1u);
}

// 3) dinv = rsqrt(deg) (in place over the u32 buffer); s init = self-loop term
__global__ void k_dinv_sinit(float* __restrict__ dinv /*aliases deg*/,
                             const float* __restrict__ x,
                             float* __restrict__ s, int n) {
    int i = blockIdx.x * blockDim.x + threadIdx.x;
    if (i >= n) return;
    unsigned d = ((const unsigned*)dinv)[i];      // deg >= 1 always (self loop)
    float di = rsqrtf((float)d);
    dinv[i] = di;
    s[i] = di * di * x[i];                        // norm_ii * x_i
}

// 4) edge pass A: s[d] += dinv[s]*dinv[d]*x[s]
__global__ void k_passA(const int* __restrict__ src, const int* __restrict__ dst,
                        const float* __restrict__ dinv, const float* __restrict__ x,
                        float* __restrict__ s, int e) {
    int i = blockIdx.x * blockDim.x + threadIdx.x;
    if (i >= e) return;
    int js = src[i], jd = dst[i];
    float w = dinv[js] * dinv[jd];
    atomicAdd(&s[jd], w * x[js]);
}

// 5) scalars c1 = W1·W2, c0 = b1·W2
__global__ void k_scal(const float* __restrict__ W1, const float* __restrict__ b1,
                       const float* __restrict__ W2, float* __restrict__ scal) {
    if (blockIdx.x == 0 && threadIdx.x == 0) {
        float c1 = 0.f, c0 = 0.f;
        for (int k = 0; k < 16; ++k) { c1 += W1[k] * W2[k]; c0 += b1[k] * W2[k]; }
        scal[0] = c1; scal[1] = c0;
    }
}

// 6) t init = self-loop term of pass B (needs s complete)
__global__ void k_tinit(const float* __restrict__ dinv, const float* __restrict__ s,
                        const float* __restrict__ scal, float* __restrict__ t, int n) {
    int i = blockIdx.x * blockDim.x + threadIdx.x;
    if (i >= n) return;
    float di = dinv[i];
    t[i] = di * di * (scal[0] * s[i] + scal[1]);
}

// 7) edge pass B: t[d] += dinv[s]*dinv[d]*(c1*s[s]+c0)
__global__ void k_passB(const int* __restrict__ src, const int* __restrict__ dst,
                        const float* __restrict__ dinv, const float* __restrict__ s,
                        const float* __restrict__ scal, float* __restrict__ t, int e) {
    int i = blockIdx.x * blockDim.x + threadIdx.x;
    if (i >= e) return;
    int js = src[i], jd = dst[i];
    float w = dinv[js] * dinv[jd];
    atomicAdd(&t[jd], w * (scal[0] * s[js] + scal[1]));
}

// ---------------------------------------------------------------------------
// 8) z = relu( (t.reshape(B,784) + b2) @ Wlin + blin ), fp32 WMMA 16x16x4.
//    One wave per 16x16 output tile. K = 784 = 196 * 4.
//    A 16x4 layout (ISA 7.12.2): lane in [0,16): holds A[M=lane][k], A[M][k+1];
//                                 lane in [16,32): A[M=lane-16][k+2], A[M][k+3].
//    B 4x16: mirror of A (N = lane%16, K selected by vgpr + lane-half).
//    C/D:    vgpr r, lane l -> C[M = r + 8*(l>>4)][N = l%16].
__global__ void k_gemm1_wmma(const float* __restrict__ t, const float* __restrict__ Wlin,
                             const float* __restrict__ blin, const float* __restrict__ b2,
                             float* __restrict__ z, int ntiles_n) {
    int wave  = blockIdx.x * (blockDim.x >> 5) + (threadIdx.x >> 5);
    int lane  = threadIdx.x & 31;
    int mt    = wave / ntiles_n;
    int nt    = wave % ntiles_n;
    int m0    = mt << 4;
    int n0    = nt << 4;
    int lmod  = lane & 15;
    int lhalf = lane >> 4;            // 0 or 1
    float b2v = b2[0];

    const float* arow = t + (size_t)(m0 + lmod) * 784;
    int acol = lhalf << 1;            // 0 or 2
    int n    = n0 + lmod;

    v8f acc = {};
    for (int k = 0; k < 784; k += 4) {
        int ka = k + acol;
        v2f a, b;
        a.x = arow[ka]     + b2v;
        a.y = arow[ka + 1] + b2v;
        b.x = Wlin[(size_t)ka       * 128 + n];
        b.y = Wlin[(size_t)(ka + 1) * 128 + n];
        acc = __builtin_amdgcn_wmma_f32_16x16x4_f32(
            /*neg_a=*/false, a, /*neg_b=*/false, b,
            /*c_mod=*/(short)0, acc, /*reuse_a=*/false, /*reuse_b=*/false);
    }
    float bias = blin[n];
    #pragma unroll
    for (int r = 0; r < 8; ++r) {
        int m = m0 + r + (lhalf << 3);
        float v = acc[r] + bias;
        z[(size_t)m * 128 + n] = v > 0.f ? v : 0.f;
    }
}

// 9) logits = z @ Wcls + bcls, then log_softmax per row (10 classes)
__global__ void k_head(const float* __restrict__ z, const float* __restrict__ Wcls,
                       const float* __restrict__ bcls, float* __restrict__ out, int nimg) {
    int m = blockIdx.x * blockDim.x + threadIdx.x;
    if (m >= nimg) return;
    const float* zr = z + (size_t)m * 128;
    float logits[10];
    #pragma unroll
    for (int n = 0; n < 10; ++n) {
        float a = bcls[n];
        for (int j = 0; j < 128; ++j) a += zr[j] * Wcls[j * 10 + n];
        logits[n] = a;
    }
    float mx = logits[0];
    #pragma unroll
    for (int n = 1; n < 10; ++n) mx = fmaxf(mx, logits[n]);
    float se = 0.f;
    #pragma unroll
    for (int n = 0; n < 10; ++n) se += expf(logits[n] - mx);
    float lse = mx + logf(se);
    #pragma unroll
    for (int n = 0; n < 10; ++n) out[(size_t)m * 10 + n] = logits[n] - lse;
}

// ---------------------------------------------------------------------------
extern "C" void kernel_launch(void* const* d_in, const int* in_sizes, int n_in,
                              void* d_out, int out_size, void* d_ws, size_t ws_size,
                              hipStream_t stream) {
    const float* x    = (const float*)d_in[0];
    const int*   ei   = (const int*)  d_in[1];
    const float* W1   = (const float*)d_in[2];
    const float* b1   = (const float*)d_in[3];
    const float* W2   = (const float*)d_in[4];
    const float* b2   = (const float*)d_in[5];
    const float* Wlin = (const float*)d_in[6];
    const float* blin = (const float*)d_in[7];
    const float* Wcls = (const float*)d_in[8];
    const float* bcls = (const float*)d_in[9];
    float* out = (float*)d_out;

    const int N = in_sizes[0];          // 1605632
    const int E = in_sizes[1] / 2;      // 12845056
    const int B = N / 784;              // 2048 images
    const int* src = ei;
    const int* dst = ei + E;

    // workspace carve-up (floats)
    float* degf = (float*)d_ws;         // N  (u32 degree, then dinv in place)
    float* sbuf = degf + N;             // N
    float* tbuf = sbuf + N;             // N
    float* zbuf = tbuf + N;             // B*128
    float* scal = zbuf + (size_t)B*128; // 2

    const int BT = 256;
    const int gN = (N + BT - 1) / BT;
    const int gE = (E + BT - 1) / BT;

    k_init_deg <<<gN, BT, 0, stream>>>((unsigned*)degf, N);
    k_count_deg<<<gE, BT, 0, stream>>>(dst, (unsigned*)degf, E);
    k_dinv_sinit<<<gN, BT, 0, stream>>>(degf, x, sbuf, N);
    k_passA    <<<gE, BT, 0, stream>>>(src, dst, degf, x, sbuf, E);
    k_scal     <<<1, 32, 0, stream>>>(W1, b1, W2, scal);
    k_tinit    <<<gN, BT, 0, stream>>>(degf, sbuf, scal, tbuf, N);
    k_passB    <<<gE, BT, 0, stream>>>(src, dst, degf, sbuf, scal, tbuf, E);

    // dense head: (B x 784) @ (784 x 128), one wave per 16x16 tile
    const int mt = B / 16;              // 128
    const int nt = 128 / 16;            // 8
    const int waves = mt * nt;          // 1024
    const int wavesPerBlk = 8;          // 256 threads
    k_gemm1_wmma<<<waves / wavesPerBlk, wavesPerBlk * 32, 0, stream>>>(
        tbuf, Wlin, blin, b2, zbuf, nt);

    k_head<<<(B + BT - 1) / BT, BT, 0, stream>>>(zbuf, Wcls, bcls, out, B);
}